// TimeDistributedReturn_7756710937364
// MI455X (gfx1250) — compile-verified
//
#include <hip/hip_runtime.h>
#include <hip/hip_bf16.h>

// time_distributed_return (batch_first=False):  R[t] = r[t] + GAMMA * R[t+1]
// Two-kernel blocked scan. The 16x16 intra-block triangular multiply AND the
// carry propagation are both done with V_WMMA_F32_16X16X4_F32 (5 WMMAs per
// 16-step sub-block: 4 for the triangular r-tile, 1 rank-1 carry update whose
// D row 0 directly feeds the next iteration's B operand - no shuffle needed).

#define T_DIM   8192
#define N_DIM   1024
#define SEG     128                  // k1 summary granularity
#define NSEG    (T_DIM / SEG)        // 64
#define CHUNK   256                  // k2 chunk (timesteps per wave)
#define NCHUNK  (T_DIM / CHUNK)      // 32
#define SUBS    (CHUNK / 16)         // 16 sub-blocks per chunk
#define GAMMA   0.99f
#define LOG2G   (-0.0144995696951151f)   // log2(0.99)

typedef float v2f __attribute__((ext_vector_type(2)));
typedef float v8f __attribute__((ext_vector_type(8)));

__device__ __forceinline__ float gpow(int e) {
    return exp2f((float)e * LOG2G);      // gamma^e via v_exp_f32
}

// ---------------------------------------------------------------------------
// k1: segment summaries  W_s[n] = sum_{t in seg} gamma^(t - s*SEG) * r[t,n]
//     (exact Horner recurrence, float4-vectorized: 4 independent chains/thread)
// ---------------------------------------------------------------------------
__global__ void __launch_bounds__(256)
k1_seg_summaries(const float* __restrict__ r, float* __restrict__ Wseg) {
    const int seg  = blockIdx.x;              // 0..NSEG-1
    const int c4   = threadIdx.x;             // 0..255 -> 4 columns each
    const int NV   = N_DIM / 4;
    const float4* r4 = (const float4*)r;

    float4 w = make_float4(0.f, 0.f, 0.f, 0.f);
    const int base = seg * SEG;
    for (int t = SEG - 1; t >= 0; --t) {
        const float4 v = r4[(base + t) * NV + c4];
        w.x = fmaf(GAMMA, w.x, v.x);
        w.y = fmaf(GAMMA, w.y, v.y);
        w.z = fmaf(GAMMA, w.z, v.z);
        w.w = fmaf(GAMMA, w.w, v.w);
    }
    ((float4*)Wseg)[seg * NV + c4] = w;
}

// ---------------------------------------------------------------------------
// k2: per (chunk, 16-col tile) wave.
//   1) carry-in G = R[chunk_end] from segment summaries (tiny Horner scan)
//   2) backward over 16-row sub-blocks:
//        D = U @ r_sub            (4x WMMA f32 16x16x4, U upper-tri gammas)
//          + a5 (x) carry         (1x WMMA rank-1: A5[m]=gamma^(16-m))
//      next carry = D row 0, already sitting in lanes 0..15 => feeds B5.x
// ---------------------------------------------------------------------------
__global__ void __launch_bounds__(128)
k2_wmma_returns(const float* __restrict__ r,
                const float* __restrict__ Wseg,
                float* __restrict__ out) {
    const int lane  = threadIdx.x & 31;
    const int wave  = threadIdx.x >> 5;
    const int ntile = blockIdx.x * 4 + wave;   // 0..63
    const int chunk = blockIdx.y;              // 0..NCHUNK-1
    const int m     = lane & 15;
    const int khalf = lane >> 4;
    const int col   = ntile * 16 + m;

    // A: U[m][j] = gamma^(j-m) (j>=m). 32-bit A 16x4 layout:
    //   VGPR0 = K(ko+2*khalf), VGPR1 = K(ko+2*khalf+1)
    v2f A[4];
#pragma unroll
    for (int kk = 0; kk < 4; ++kk) {
        const int j0 = kk * 4 + 2 * khalf;
        A[kk].x = (j0     >= m) ? gpow(j0     - m) : 0.f;
        A[kk].y = (j0 + 1 >= m) ? gpow(j0 + 1 - m) : 0.f;
    }
    // A5: rank-1 carry column, K=0 only => lanes 0..15 of VGPR0
    v2f A5;
    A5.x = (khalf == 0) ? gpow(16 - m) : 0.f;
    A5.y = 0.f;

    // carry-in: G = sum_{s >= first seg after chunk} gamma^((s-s0)*SEG) * W_s
    const float gS = gpow(SEG);
    float g = 0.f;
    const int sfirst = (chunk + 1) * (CHUNK / SEG);
    for (int s = NSEG - 1; s >= sfirst; --s)
        g = fmaf(gS, g, Wseg[s * N_DIM + col]);

    // B5: carry in B layout, K=0 row only (lanes 0..15 of VGPR0)
    v2f B5;
    B5.x = (khalf == 0) ? g : 0.f;
    B5.y = 0.f;

    const int base = chunk * CHUNK;

    // preload last sub-block (B mirrors A's K layout)
    v2f B[4];
    {
        const int t0 = base + (SUBS - 1) * 16;
#pragma unroll
        for (int kk = 0; kk < 4; ++kk) {
            const int row = t0 + kk * 4 + 2 * khalf;
            B[kk].x = r[row * N_DIM + col];
            B[kk].y = r[(row + 1) * N_DIM + col];
        }
    }

    for (int sb = SUBS - 1; sb >= 0; --sb) {
        v2f Bn[4];
        if (sb > 0) {
            const int t0n = base + (sb - 1) * 16;
#pragma unroll
            for (int kk = 0; kk < 4; ++kk) {
                const int row = t0n + kk * 4 + 2 * khalf;
                Bn[kk].x = r[row * N_DIM + col];
                Bn[kk].y = r[(row + 1) * N_DIM + col];
            }
        }

        // carry WMMA first (seeds accumulator), then the 4 triangular steps
        v8f acc = __builtin_amdgcn_wmma_f32_16x16x4_f32(
            false, A5, false, B5, (short)0, (v8f){}, false, false);
#pragma unroll
        for (int kk = 0; kk < 4; ++kk)
            acc = __builtin_amdgcn_wmma_f32_16x16x4_f32(
                false, A[kk], false, B[kk], (short)0, acc, false, false);

        const int t0 = base + sb * 16;
#pragma unroll
        for (int k = 0; k < 8; ++k)
            out[(t0 + k + 8 * khalf) * N_DIM + col] = acc[k];

        // next carry = D row 0 (lanes 0..15 hold M=0) -> directly into B5
        B5.x = (khalf == 0) ? acc[0] : 0.f;

#pragma unroll
        for (int kk = 0; kk < 4; ++kk) B[kk] = Bn[kk];
    }
}

extern "C" void kernel_launch(void* const* d_in, const int* in_sizes, int n_in,
                              void* d_out, int out_size, void* d_ws, size_t ws_size,
                              hipStream_t stream) {
    const float* r   = (const float*)d_in[0];
    float*       out = (float*)d_out;
    float*       Ws  = (float*)d_ws;           // NSEG * N_DIM floats (256 KB)

    k1_seg_summaries<<<NSEG, 256, 0, stream>>>(r, Ws);

    dim3 g2(16, NCHUNK);                       // 16 blocks * 4 waves = 64 col tiles
    k2_wmma_returns<<<g2, 128, 0, stream>>>(r, Ws, out);
}